// RPNPostProcessor_71330816851991
// MI455X (gfx1250) — compile-verified
//
#include <hip/hip_runtime.h>
#include <hip/hip_bf16.h>
#include <cstdint>

// ---------------- problem constants (from reference) ----------------
#define N_IMG   16
#define A_ANC   3
#define H_DIM   256
#define W_DIM   256
#define HW_DIM  (H_DIM * W_DIM)            // 65536
#define M_CAND  (A_ANC * HW_DIM)           // 196608
#define K_PRE   2000                       // PRE_NMS_TOP_N
#define K_PAD   2048                       // padded for bitonic / scans
#define K_CAP   4096                       // compact capacity (select bin ties)
#define K_POST  1000                       // POST_NMS_TOP_N
#define NMS_T   0.7f
#define BBOX_CLIP 4.135166556742356f       // log(1000/16)

// ---------------- workspace layout (bytes) ----------------
#define OFF_HIST   0u                      // N*2048 u32      = 131072
#define OFF_TC     131072u                 // N*2 i32  (T,C1)
#define OFF_CNT    131328u                 // N*2 u32  (atomic counters)
#define OFF_KEYS   131584u                 // N*4096 u64      = 524288
#define OFF_BOXES  655872u                 // N*2048*4 f32    = 524288
#define OFF_SCORES 1180160u                // N*2048 f32      = 131072
#define OFF_KEEP   1311232u                // N*2048 u32      = 131072
#define ZERO_WORDS ((OFF_KEYS + 524288u) / 4u)   // zero hist..keys each call

// float bits -> order-preserving unsigned (ascending)
__device__ __forceinline__ unsigned ordbits(unsigned b) {
    return b ^ ((b & 0x80000000u) ? 0xFFFFFFFFu : 0x80000000u);
}
__device__ __forceinline__ unsigned unordbits(unsigned u) {
    return (u & 0x80000000u) ? (u ^ 0x80000000u) : ~u;
}

__global__ void k_zero(unsigned* __restrict__ p, int nwords) {
    int i = blockIdx.x * blockDim.x + threadIdx.x;
    if (i < nwords) p[i] = 0u;
}

// Pass 1: histogram of top-11 ordered logit bits, per image.
// gid order = (n, a, p) so objectness reads are fully coalesced.
__global__ void k_hist(const float* __restrict__ obj, unsigned* __restrict__ hist) {
    int gid = blockIdx.x * blockDim.x + threadIdx.x;
    if (gid >= N_IMG * M_CAND) return;
    int n = gid / M_CAND;
    int r = gid - n * M_CAND;                 // r = a*HW + p
    __builtin_prefetch(&obj[gid + 8192], 0, 1);   // global_prefetch_b8
    float logit = obj[gid];
    unsigned u = ordbits(__float_as_uint(logit));
    atomicAdd(&hist[n * 2048 + (u >> 21)], 1u);
    (void)r;
}

// Pass 2: find threshold bin T (2000th element lives in it) + count above (C1).
__global__ void k_thresh(const unsigned* __restrict__ hist, int* __restrict__ tc) {
    int n = blockIdx.x * blockDim.x + threadIdx.x;
    if (n >= N_IMG) return;
    unsigned acc = 0; int T = 0; unsigned c1 = 0;
    for (int b = 2047; b >= 0; --b) {
        unsigned h = hist[n * 2048 + b];
        if (acc + h >= (unsigned)K_PRE) { T = b; c1 = acc; break; }
        acc += h;
        if (b == 0) { T = 0; c1 = acc; }
    }
    tc[2 * n] = T; tc[2 * n + 1] = (int)c1;
}

// Pass 3: compact candidates with bin >= T. key = ord(score)<<32 | ~idx
// (descending score, ties -> smaller original index first, exactly like top_k).
__global__ void k_compact(const float* __restrict__ obj, const int* __restrict__ tc,
                          unsigned* __restrict__ cnt, unsigned long long* __restrict__ keys) {
    int gid = blockIdx.x * blockDim.x + threadIdx.x;
    if (gid >= N_IMG * M_CAND) return;
    int n = gid / M_CAND;
    int r = gid - n * M_CAND;
    int a = r / HW_DIM;
    int p = r - a * HW_DIM;
    unsigned m = (unsigned)(p * A_ANC + a);   // flattened (h,w,a) candidate index
    float logit = obj[gid];
    unsigned u = ordbits(__float_as_uint(logit));
    int bin = (int)(u >> 21);
    int T = tc[2 * n];
    unsigned C1 = (unsigned)tc[2 * n + 1];
    if (bin < T) return;
    unsigned long long key = ((unsigned long long)u << 32) | (unsigned)(~m);
    unsigned slot;
    if (bin > T) slot = atomicAdd(&cnt[2 * n], 1u);
    else         slot = C1 + atomicAdd(&cnt[2 * n + 1], 1u);
    if (slot < K_CAP) keys[(size_t)n * K_CAP + slot] = key;
}

// Pass 4: per-image bitonic sort (descending) of 4096 keys, entirely in LDS.
__global__ __launch_bounds__(1024) void k_sort(unsigned long long* __restrict__ keys) {
    __shared__ unsigned long long sk[K_CAP];   // 32 KB of the 320 KB WGP LDS
    int n = blockIdx.x;
    for (int i = threadIdx.x; i < K_CAP; i += 1024) sk[i] = keys[(size_t)n * K_CAP + i];
    __syncthreads();
    for (int k = 2; k <= K_CAP; k <<= 1) {
        for (int j = k >> 1; j > 0; j >>= 1) {
            for (int i = threadIdx.x; i < K_CAP; i += 1024) {
                int ixj = i ^ j;
                if (ixj > i) {
                    unsigned long long x = sk[i], y = sk[ixj];
                    bool down = ((i & k) == 0);           // descending overall
                    if (down ? (x < y) : (x > y)) { sk[i] = y; sk[ixj] = x; }
                }
            }
            __syncthreads();
        }
    }
    for (int i = threadIdx.x; i < K_PAD; i += 1024) keys[(size_t)n * K_CAP + i] = sk[i];
}

// Pass 5: gather regression+anchors for top-2000, decode, clip, validity.
__global__ void k_decode(const unsigned long long* __restrict__ keys,
                         const float* __restrict__ br, const float* __restrict__ anc,
                         const int* __restrict__ ishape,
                         float* __restrict__ boxes, float* __restrict__ scores,
                         unsigned* __restrict__ keep) {
    int gid = blockIdx.x * blockDim.x + threadIdx.x;
    if (gid >= N_IMG * K_PAD) return;
    int n = gid / K_PAD;
    int k = gid - n * K_PAD;
    unsigned long long key = keys[(size_t)n * K_CAP + k];
    float bx1 = 0.f, by1 = 0.f, bx2 = 0.f, by2 = 0.f, sc = 0.f;
    unsigned kp = 0u;
    if (k < K_PRE && key != 0ull) {
        unsigned u = (unsigned)(key >> 32);
        unsigned m = ~(unsigned)(key & 0xFFFFFFFFull);
        float logit = __uint_as_float(unordbits(u));
        sc = 1.f / (1.f + expf(-logit));
        int a = (int)(m % A_ANC);
        int p = (int)(m / A_ANC);
        const float* an = &anc[((size_t)n * M_CAND + m) * 4];
        float ax1 = an[0], ay1 = an[1], ax2 = an[2], ay2 = an[3];
        float wa = ax2 - ax1 + 1.f, ha = ay2 - ay1 + 1.f;
        float cxa = ax1 + 0.5f * wa, cya = ay1 + 0.5f * ha;
        size_t base = (size_t)n * (A_ANC * 4 * HW_DIM);
        float dx = br[base + (size_t)(a * 4 + 0) * HW_DIM + p];
        float dy = br[base + (size_t)(a * 4 + 1) * HW_DIM + p];
        float dw = fminf(br[base + (size_t)(a * 4 + 2) * HW_DIM + p], BBOX_CLIP);
        float dh = fminf(br[base + (size_t)(a * 4 + 3) * HW_DIM + p], BBOX_CLIP);
        float cx = dx * wa + cxa, cy = dy * ha + cya;
        float w = expf(dw) * wa,  h = expf(dh) * ha;
        bx1 = cx - 0.5f * w; by1 = cy - 0.5f * h;
        bx2 = cx + 0.5f * w - 1.f; by2 = cy + 0.5f * h - 1.f;
        float wim = (float)ishape[2 * n + 0] - 1.f;
        float him = (float)ishape[2 * n + 1] - 1.f;
        bx1 = fminf(fmaxf(bx1, 0.f), wim); bx2 = fminf(fmaxf(bx2, 0.f), wim);
        by1 = fminf(fmaxf(by1, 0.f), him); by2 = fminf(fmaxf(by2, 0.f), him);
        float ws_ = bx2 - bx1 + 1.f, hs_ = by2 - by1 + 1.f;
        kp = (ws_ >= 0.f && hs_ >= 0.f) ? 1u : 0u;   // MIN_SIZE == 0
    }
    float* bp = &boxes[(size_t)gid * 4];
    bp[0] = bx1; bp[1] = by1; bp[2] = bx2; bp[3] = by2;
    scores[gid] = sc;
    keep[gid] = kp;
}

// Pass 6: greedy NMS, one workgroup (32 wave32s) per image, all operands in LDS.
// Boxes staged via CDNA5 async global->LDS DMA (ASYNCcnt path).
__global__ __launch_bounds__(1024) void k_nms(const float* __restrict__ boxes_g,
                                              unsigned* __restrict__ keep_g) {
    __shared__ float    sb[K_PAD * 4];     // 32 KB
    __shared__ float    sarea[K_PAD];      //  8 KB
    __shared__ unsigned skeep[K_PAD];      //  8 KB
    int n = blockIdx.x;
    const float* src = boxes_g + (size_t)n * K_PAD * 4;
    unsigned long long sbase = (unsigned long long)(uintptr_t)src;
    unsigned lds0 = (unsigned)(uintptr_t)(&sb[0]);
    for (int k = threadIdx.x; k < K_PAD; k += 1024) {
        unsigned voff = (unsigned)k * 16u;
        unsigned vlds = lds0 + voff;
        // gfx1250 async DMA: copies 16 bytes from global base sbase plus offset voff to the LDS address in vlds, tracked by ASYNCcnt
        asm volatile("global_load_async_to_lds_b128 %0, %1, %2"
                     :: "v"(vlds), "v"(voff), "s"(sbase) : "memory");
    }
    asm volatile("s_wait_asynccnt 0x0" ::: "memory");
    __syncthreads();
    for (int k = threadIdx.x; k < K_PAD; k += 1024) {
        skeep[k] = keep_g[(size_t)n * K_PAD + k];
        float x1 = sb[4 * k], y1 = sb[4 * k + 1], x2 = sb[4 * k + 2], y2 = sb[4 * k + 3];
        sarea[k] = (x2 - x1 + 1.f) * (y2 - y1 + 1.f);
    }
    __syncthreads();
    for (int i = 0; i < K_PRE; ++i) {
        if (skeep[i]) {                           // uniform across the block
            float xi1 = sb[4 * i], yi1 = sb[4 * i + 1];
            float xi2 = sb[4 * i + 2], yi2 = sb[4 * i + 3];
            float ai = sarea[i];
            for (int j = i + 1 + threadIdx.x; j < K_PRE; j += 1024) {
                float iw = fminf(xi2, sb[4 * j + 2]) - fmaxf(xi1, sb[4 * j]) + 1.f;
                float ih = fminf(yi2, sb[4 * j + 3]) - fmaxf(yi1, sb[4 * j + 1]) + 1.f;
                iw = fmaxf(iw, 0.f); ih = fmaxf(ih, 0.f);
                float inter = iw * ih;
                float iou = inter / (ai + sarea[j] - inter);
                if (iou > NMS_T) skeep[j] = 0u;
            }
        }
        __syncthreads();
    }
    for (int k = threadIdx.x; k < K_PAD; k += 1024) keep_g[(size_t)n * K_PAD + k] = skeep[k];
}

// Pass 7: stable compaction of survivors -> first 1000 rows of (boxes|score).
__global__ __launch_bounds__(1024) void k_select(const float* __restrict__ boxes_g,
                                                 const float* __restrict__ scores_g,
                                                 const unsigned* __restrict__ keep_g,
                                                 float* __restrict__ out) {
    __shared__ unsigned ssum[K_PAD];
    int n = blockIdx.x;
    int t = threadIdx.x;
    for (int k = t; k < K_PAD; k += 1024)
        ssum[k] = (k < K_PRE) ? keep_g[(size_t)n * K_PAD + k] : 0u;
    __syncthreads();
    // Hillis-Steele inclusive scan over 2048 (2 elems/thread)
    for (int off = 1; off < K_PAD; off <<= 1) {
        int k0 = t, k1 = t + 1024;
        unsigned v0 = (k0 >= off) ? ssum[k0 - off] : 0u;
        unsigned v1 = (k1 >= off) ? ssum[k1 - off] : 0u;
        __syncthreads();
        ssum[k0] += v0; ssum[k1] += v1;
        __syncthreads();
    }
    for (int k = t; k < K_PRE; k += 1024) {
        if (keep_g[(size_t)n * K_PAD + k]) {
            unsigned rnk = ssum[k] - 1u;
            if (rnk < K_POST) {
                float* o = &out[((size_t)n * K_POST + rnk) * 5];
                const float* b = &boxes_g[(size_t)((size_t)n * K_PAD + k) * 4];
                o[0] = b[0]; o[1] = b[1]; o[2] = b[2]; o[3] = b[3];
                o[4] = scores_g[(size_t)n * K_PAD + k];
            }
        }
    }
}

extern "C" void kernel_launch(void* const* d_in, const int* in_sizes, int n_in,
                              void* d_out, int out_size, void* d_ws, size_t ws_size,
                              hipStream_t stream) {
    (void)in_sizes; (void)n_in; (void)ws_size;
    const float* obj = (const float*)d_in[0];
    const float* br  = (const float*)d_in[1];
    const float* anc = (const float*)d_in[2];
    const int*   ish = (const int*)d_in[3];
    float* out = (float*)d_out;
    char*  ws  = (char*)d_ws;

    unsigned*           hist   = (unsigned*)(ws + OFF_HIST);
    int*                tc     = (int*)(ws + OFF_TC);
    unsigned*           cnt    = (unsigned*)(ws + OFF_CNT);
    unsigned long long* keys   = (unsigned long long*)(ws + OFF_KEYS);
    float*              boxes  = (float*)(ws + OFF_BOXES);
    float*              scores = (float*)(ws + OFF_SCORES);
    unsigned*           keep   = (unsigned*)(ws + OFF_KEEP);

    // zero histograms, counters, key slots, and the output each call
    k_zero<<<((int)ZERO_WORDS + 255) / 256, 256, 0, stream>>>((unsigned*)ws, (int)ZERO_WORDS);
    k_zero<<<(out_size + 255) / 256, 256, 0, stream>>>((unsigned*)d_out, out_size);

    int total = N_IMG * M_CAND;
    k_hist   <<<(total + 255) / 256, 256, 0, stream>>>(obj, hist);
    k_thresh <<<1, 32, 0, stream>>>(hist, tc);
    k_compact<<<(total + 255) / 256, 256, 0, stream>>>(obj, tc, cnt, keys);
    k_sort   <<<N_IMG, 1024, 0, stream>>>(keys);
    k_decode <<<(N_IMG * K_PAD + 255) / 256, 256, 0, stream>>>(keys, br, anc, ish,
                                                               boxes, scores, keep);
    k_nms    <<<N_IMG, 1024, 0, stream>>>(boxes, keep);
    k_select <<<N_IMG, 1024, 0, stream>>>(boxes, scores, keep, out);
}